// RewireGearnet_781684048169
// MI455X (gfx1250) — compile-verified
//
#include <hip/hip_runtime.h>

// ---------------- problem constants (match reference) ----------------
#define NNODES 10000
#define EEDGES 160000
#define DDIM   512
#define RNUM   7
#define LNUM   3
#define KSPLIT (RNUM * DDIM)        // 3584 : boundary between upd-part and h-part of K
#define KTOT   (KSPLIT + DDIM)      // 4096 : fused GEMM K

// ---------------- GEMM tiling ----------------
#define BM 64
#define BN 64
#define BK 32
#define SA 36   // LDS row stride for A tile: mult-of-4 (16B async chunks), conflict-free
#define SB 80   // LDS row stride for B tile: 2*80 mod 64 == 32 -> disjoint bank windows

typedef float v2f __attribute__((ext_vector_type(2)));
typedef float v8f __attribute__((ext_vector_type(8)));

// =====================================================================
// 1) zero workspace (float4 stores)
// =====================================================================
__global__ void zero_f4_kernel(float4* __restrict__ p, long n4) {
    long i = (long)blockIdx.x * blockDim.x + threadIdx.x;
    long stride = (long)gridDim.x * blockDim.x;
    float4 z = {0.f, 0.f, 0.f, 0.f};
    for (; i < n4; i += stride) p[i] = z;
}

// =====================================================================
// 2) edge scatter: upd[(node_out*R + rel)*D + d] += w * h[node_in*D + d]
//    one 128-thread block per edge, 4 floats per thread
// =====================================================================
__global__ void scatter_kernel(const float* __restrict__ h,
                               const float* __restrict__ ew,
                               const int*   __restrict__ nin,
                               const int*   __restrict__ nout,
                               const int*   __restrict__ rel,
                               float*       __restrict__ upd) {
    int e = blockIdx.x;
    int t = threadIdx.x;                       // 0..127
    int   src = nin[e];
    long  seg = (long)nout[e] * RNUM + rel[e];
    float w   = ew[e];
    const float4* hs = (const float4*)(h + (long)src * DDIM);
    float4 v = hs[t];                          // h row is L2-resident (20MB << 192MB)
    float* dst = upd + seg * DDIM + t * 4;
    unsafeAtomicAdd(dst + 0, w * v.x);
    unsafeAtomicAdd(dst + 1, w * v.y);
    unsafeAtomicAdd(dst + 2, w * v.z);
    unsafeAtomicAdd(dst + 3, w * v.w);
}

// =====================================================================
// 3) fused GEMM + bias + ReLU via V_WMMA_F32_16X16X4_F32 (full fp32)
//    out[m, n] = relu( sum_k A[m,k] W[k,n] + bl[n] + bs[n] )
//    A[m, 0:3584]   = upd row m   (ld = 3584); A[m, 3584:4096] = hin row m (ld = 512)
//    W[0:3584, n]   = W_lin (ld = 512);        W[3584:4096, n] = W_self   (ld = 512)
//    Block: 256 thr = 8 waves; block tile 64x64; wave tile 16x32 (2 accums)
//    Tiles staged via async DMA (global_load_async_to_lds_b128), double-buffered.
// =====================================================================
__global__ __launch_bounds__(256)
void gemm_fused_kernel(const float* __restrict__ Aupd,
                       const float* __restrict__ Ah,
                       const float* __restrict__ Wlin,
                       const float* __restrict__ Wself,
                       const float* __restrict__ blin,
                       const float* __restrict__ bself,
                       float*       __restrict__ outp,
                       int M) {
    __shared__ float sA[2][BM * SA];   // 2 x 9216 B
    __shared__ float sB[2][BK * SB];   // 2 x 10240 B

    const int t   = threadIdx.x;
    const int m0  = blockIdx.x * BM;
    const int n0  = blockIdx.y * BN;
    const int w   = t >> 5;            // wave 0..7
    const int wm  = w & 3;             // row-wave 0..3  (16 rows each)
    const int wn  = w >> 2;            // col-wave 0..1  (32 cols each)
    const int l   = t & 31;            // lane
    const int m   = l & 15;            // M (or N) index within fragment
    const int kb2 = (l >> 4) * 2;      // K sub-base per half-wave (0 or 2)

    v8f c0 = {};
    v8f c1 = {};

    const int nkt = KTOT / BK;         // 128 K-tiles

    // issue async DMA of K-tile `kt` into LDS buffer `buf` (4 x 16B per thread)
    auto issue_tile = [&](int kt, int buf) {
        const int k0 = kt * BK;
        const bool first = (k0 < KSPLIT);
        const float* __restrict__ Asrc = first ? Aupd : Ah;
        const int lda   = first ? KSPLIT : DDIM;
        const int kbase = first ? k0 : (k0 - KSPLIT);
        const float* __restrict__ Wsrc = first ? Wlin : Wself;
        // A tile: 64 rows x 32 cols, 16B chunks, 8 chunks/row, coalesced
        #pragma unroll
        for (int i = 0; i < 2; ++i) {
            int q  = t + i * 256;            // chunk id 0..511
            int r  = q >> 3;                 // 0..63
            int c4 = (q & 7) * 4;            // 0,4,..,28
            int rg = m0 + r; if (rg >= M) rg = M - 1;   // clamp: pad rows never stored
            const void* g = (const void*)(Asrc + (long)rg * lda + kbase + c4);
            unsigned    d = (unsigned)(uintptr_t)&sA[buf][r * SA + c4];
            asm volatile("global_load_async_to_lds_b128 %0, %1, off"
                         :: "v"(d), "v"(g) : "memory");
        }
        // B tile: 32 rows(K) x 64 cols(N), 16B chunks, 16 chunks/row, coalesced
        #pragma unroll
        for (int i = 0; i < 2; ++i) {
            int q  = t + i * 256;            // chunk id 0..511
            int r  = q >> 4;                 // 0..31
            int c4 = (q & 15) * 4;           // 0,4,..,60
            const void* g = (const void*)(Wsrc + (long)(kbase + r) * DDIM + n0 + c4);
            unsigned    d = (unsigned)(uintptr_t)&sB[buf][r * SB + c4];
            asm volatile("global_load_async_to_lds_b128 %0, %1, off"
                         :: "v"(d), "v"(g) : "memory");
        }
    };

    // prologue: DMA tile 0, wait, barrier
    issue_tile(0, 0);
    asm volatile("s_wait_asynccnt 0x0" ::: "memory");
    __syncthreads();

    for (int kt = 0; kt < nkt; ++kt) {
        const int cur = kt & 1;
        if (kt + 1 < nkt) issue_tile(kt + 1, cur ^ 1);   // DMA next tile under compute

        // ---- 8 WMMA K-steps of 4 on current buffer ----
        #pragma unroll
        for (int kk = 0; kk < 8; ++kk) {
            const int k = kk * 4;
            v2f a, b0, b1;
            // A 16x4 fragment: lane m holds A[m][k+kb2 .. +1]
            a.x = sA[cur][(wm * 16 + m) * SA + (k + kb2)];
            a.y = sA[cur][(wm * 16 + m) * SA + (k + kb2 + 1)];
            // B 4x16 fragments: lane m holds W[k+kb2 ..][nsub + m]
            const int nsub0 = wn * 32 + m;
            b0.x = sB[cur][(k + kb2) * SB + nsub0];
            b0.y = sB[cur][(k + kb2 + 1) * SB + nsub0];
            b1.x = sB[cur][(k + kb2) * SB + nsub0 + 16];
            b1.y = sB[cur][(k + kb2 + 1) * SB + nsub0 + 16];
            c0 = __builtin_amdgcn_wmma_f32_16x16x4_f32(false, a, false, b0,
                                                       (short)0, c0, false, false);
            c1 = __builtin_amdgcn_wmma_f32_16x16x4_f32(false, a, false, b1,
                                                       (short)0, c1, false, false);
        }

        // DMA for next tile must land before anyone reads it
        asm volatile("s_wait_asynccnt 0x0" ::: "memory");
        __syncthreads();
    }

    // ---- epilogue: bias + relu + store ----
    const int cg0 = n0 + wn * 32 + m;
    const int cg1 = cg0 + 16;
    const float bias0 = blin[cg0] + bself[cg0];
    const float bias1 = blin[cg1] + bself[cg1];
    const int rbase = wm * 16 + (l >> 4) * 8;   // C/D layout: VGPR r -> row r / r+8
    #pragma unroll
    for (int r = 0; r < 8; ++r) {
        int rg = m0 + rbase + r;
        if (rg < M) {
            float v0 = c0[r] + bias0; v0 = v0 > 0.f ? v0 : 0.f;
            float v1 = c1[r] + bias1; v1 = v1 > 0.f ? v1 : 0.f;
            outp[(long)rg * DDIM + cg0] = v0;
            outp[(long)rg * DDIM + cg1] = v1;
        }
    }
}

// =====================================================================
// 4) graph readout: column sums (coalesced: thread c reads h[r*D + c])
// =====================================================================
__global__ void colsum_kernel(const float* __restrict__ h, float* __restrict__ outp) {
    int c = blockIdx.x * blockDim.x + threadIdx.x;
    if (c >= DDIM) return;
    float s = 0.f;
    for (int r = 0; r < NNODES; ++r) s += h[(long)r * DDIM + c];
    outp[c] = s;
}

// =====================================================================
// launch
// =====================================================================
extern "C" void kernel_launch(void* const* d_in, const int* in_sizes, int n_in,
                              void* d_out, int out_size, void* d_ws, size_t ws_size,
                              hipStream_t stream) {
    (void)in_sizes; (void)n_in; (void)out_size; (void)ws_size;

    const float* x      = (const float*)d_in[0];
    const float* ew     = (const float*)d_in[1];
    const float* W_lin  = (const float*)d_in[2];  // [L, 3584, 512]
    const float* b_lin  = (const float*)d_in[3];  // [L, 512]
    const float* W_self = (const float*)d_in[4];  // [L, 512, 512]
    const float* b_self = (const float*)d_in[5];  // [L, 512]
    const int* node_in  = (const int*)d_in[6];
    const int* node_out = (const int*)d_in[7];
    const int* relation = (const int*)d_in[8];
    float* out = (float*)d_out;                   // [512 graph | 10000*512 node]

    float* upd = (float*)d_ws;                               // N*R*D floats
    float* h1  = upd + (size_t)NNODES * RNUM * DDIM;         // N*D
    float* h2  = h1 + (size_t)NNODES * DDIM;                 // N*D

    const long updN  = (long)NNODES * RNUM * DDIM;           // 35,840,000
    const dim3 gemmGrid((NNODES + BM - 1) / BM, DDIM / BN);  // (157, 8)

    const float* hin = x;
    for (int layer = 0; layer < LNUM; ++layer) {
        zero_f4_kernel<<<4096, 256, 0, stream>>>((float4*)upd, updN / 4);
        scatter_kernel<<<EEDGES, 128, 0, stream>>>(hin, ew, node_in, node_out,
                                                   relation, upd);
        float* hout = (layer == 0) ? h1 : (layer == 1) ? h2 : (out + DDIM);
        gemm_fused_kernel<<<gemmGrid, 256, 0, stream>>>(
            upd, hin,
            W_lin  + (size_t)layer * KSPLIT * DDIM,
            W_self + (size_t)layer * DDIM * DDIM,
            b_lin  + (size_t)layer * DDIM,
            b_self + (size_t)layer * DDIM,
            hout, NNODES);
        hin = hout;
    }
    colsum_kernel<<<2, 256, 0, stream>>>(out + DDIM, out);
}